// PinnNs_3032246911612
// MI455X (gfx1250) — compile-verified
//
#include <hip/hip_runtime.h>
#include <hip/hip_bf16.h>

// PINN Navier-Stokes via 13-component Taylor-jet forward propagation.
// Hidden-layer jet GEMMs run on V_WMMA_F32_16X16X4_F32 (f32 matrix pipe,
// K=20 = 5 exact chunks of 4). One wave handles 16 collocation points.

#define NJ   13   // jet components: 1 val + 3 first + 5 second + 4 third
#define H    20   // hidden width
#define NHID 7    // hidden->hidden layers
#define TP   16   // points per wave (WMMA M dimension)

typedef float v2f __attribute__((ext_vector_type(2)));
typedef float v8f __attribute__((ext_vector_type(8)));

// jet indices: 0:val 1:x 2:y 3:t 4:xx 5:xy 6:yy 7:xt 8:yt 9:xxx 10:xxy 11:xyy 12:yyy
__device__ __forceinline__ void jet_tanh(const float z[NJ], float h[NJ]) {
    float f  = tanhf(z[0]);
    float f1 = 1.0f - f * f;          // f'
    float f2 = -2.0f * f * f1;        // f''
    float f3 = -2.0f * f1 * f1 - 2.0f * f * f2;  // f'''
    h[0] = f;
    h[1] = f1 * z[1];
    h[2] = f1 * z[2];
    h[3] = f1 * z[3];
    h[4] = f2 * z[1] * z[1] + f1 * z[4];
    h[5] = f2 * z[1] * z[2] + f1 * z[5];
    h[6] = f2 * z[2] * z[2] + f1 * z[6];
    h[7] = f2 * z[1] * z[3] + f1 * z[7];
    h[8] = f2 * z[2] * z[3] + f1 * z[8];
    // Faa di Bruno, 3rd order
    h[9]  = f3*z[1]*z[1]*z[1] + 3.0f*f2*z[4]*z[1]                    + f1*z[9];
    h[10] = f3*z[1]*z[1]*z[2] + f2*(z[4]*z[2] + 2.0f*z[5]*z[1])      + f1*z[10];
    h[11] = f3*z[1]*z[2]*z[2] + f2*(z[6]*z[1] + 2.0f*z[5]*z[2])      + f1*z[11];
    h[12] = f3*z[2]*z[2]*z[2] + 3.0f*f2*z[6]*z[2]                    + f1*z[12];
}

__global__ __launch_bounds__(32) void pinn_ns_wmma(
    const float* __restrict__ x, const float* __restrict__ y, const float* __restrict__ t,
    const float* __restrict__ W_in, const float* __restrict__ b_in,
    const float* __restrict__ W_hid, const float* __restrict__ b_hid,
    const float* __restrict__ W_out, const float* __restrict__ b_out,
    const float* __restrict__ lb, const float* __restrict__ ub,
    const float* __restrict__ lam1, const float* __restrict__ lam2,
    float* __restrict__ out, int n)
{
    // double-buffered activation jets: [buf][jet][neuron][point]
    __shared__ float hbuf[2][NJ][H][TP];   // 33280 B
    __shared__ float outj[2][NJ][TP];      // psi / p output jets

    const int lane  = threadIdx.x;      // 0..31, single wave per block
    const int base  = blockIdx.x * TP;
    const int nloc  = lane & 15;
    const int khalf = lane >> 4;

    // ---------------- Phase 0: input layer (K=3, sparse jets) ----------------
    const float lb0 = lb[0], lb1 = lb[1], lb2 = lb[2];
    const float i0 = 1.0f / (ub[0] - lb0);
    const float i1 = 1.0f / (ub[1] - lb1);
    const float i2 = 1.0f / (ub[2] - lb2);
    #pragma unroll
    for (int it = 0; it < (H * TP) / 32; ++it) {   // 20*16 elems, 10 per lane
        int e  = it * 32 + lane;
        int nn = e >> 4;            // neuron 0..19
        int p  = e & 15;            // point  0..15
        int g  = base + p; if (g >= n) g = n - 1;
        float hx = 2.0f * (x[g] - lb0) * i0 - 1.0f;
        float hy = 2.0f * (y[g] - lb1) * i1 - 1.0f;
        float ht = 2.0f * (t[g] - lb2) * i2 - 1.0f;
        float w0 = W_in[0 * H + nn], w1 = W_in[1 * H + nn], w2 = W_in[2 * H + nn];
        float z[NJ], h[NJ];
        #pragma unroll
        for (int j = 0; j < NJ; ++j) z[j] = 0.0f;
        z[0] = hx * w0 + hy * w1 + ht * w2 + b_in[nn];
        z[1] = 2.0f * i0 * w0;      // input is affine in (x,y,t): only 1st-order jets
        z[2] = 2.0f * i1 * w1;
        z[3] = 2.0f * i2 * w2;
        jet_tanh(z, h);
        #pragma unroll
        for (int j = 0; j < NJ; ++j) hbuf[0][j][nn][p] = h[j];
    }
    __syncthreads();

    // ---------------- Phase 1: 7 hidden layers on f32 WMMA ----------------
    for (int l = 0; l < NHID; ++l) {
        const float* W  = W_hid + l * H * H;   // row-major [20][20]
        const float* bb = b_hid + l * H;
        float (*cur)[H][TP] = hbuf[l & 1];
        float (*nxt)[H][TP] = hbuf[(l + 1) & 1];
        #pragma unroll
        for (int tile = 0; tile < 2; ++tile) {          // neurons 0-15, 16-19(+pad)
            const int  nglob = tile * 16 + nloc;
            const bool valid = nglob < H;
            // B fragments (4x16 per K-chunk): VGPR r, lane -> K=4kc+r+2*khalf, N=nloc
            v2f bfrag[5];
            #pragma unroll
            for (int kc = 0; kc < 5; ++kc) {
                #pragma unroll
                for (int r = 0; r < 2; ++r) {
                    int kk = kc * 4 + r + 2 * khalf;
                    bfrag[kc][r] = valid ? W[kk * H + nglob] : 0.0f;
                }
            }
            // 13 jet GEMMs share B fragments; accumulators stay live so the
            // tanh-jet combine is lane-local (identical fragment layouts).
            v8f c[NJ];
            #pragma unroll
            for (int j = 0; j < NJ; ++j) {
                v8f acc = {};
                #pragma unroll
                for (int kc = 0; kc < 5; ++kc) {        // K = 20 = 5 x 4, exact
                    v2f a;                              // A: M=nloc, K=4kc+r+2*khalf
                    int kk0 = kc * 4 + 2 * khalf;
                    a[0] = cur[j][kk0 + 0][nloc];
                    a[1] = cur[j][kk0 + 1][nloc];
                    acc = __builtin_amdgcn_wmma_f32_16x16x4_f32(
                        false, a, false, bfrag[kc], (short)0, acc, false, false);
                }
                c[j] = acc;
            }
            // combine: lane holds all 13 z-jets of (point m, neuron nglob)
            const float bv = valid ? bb[nglob] : 0.0f;
            #pragma unroll
            for (int r = 0; r < 8; ++r) {
                int m = r + 8 * khalf;                  // C layout: M = r + 8*(lane>>4)
                float z[NJ], h[NJ];
                #pragma unroll
                for (int j = 0; j < NJ; ++j) z[j] = c[j][r];
                z[0] += bv;
                jet_tanh(z, h);
                if (valid) {
                    #pragma unroll
                    for (int j = 0; j < NJ; ++j) nxt[j][nglob][m] = h[j];
                }
            }
        }
        __syncthreads();
    }

    // ---------------- Phase 2: output layer (20 -> 2), elementwise ----------------
    {
        float (*cur)[H][TP] = hbuf[NHID & 1];   // layer-7 activations live in hbuf[1]
        int col = lane >> 4;                    // 0: psi, 1: p
        int p   = lane & 15;
        #pragma unroll
        for (int j = 0; j < NJ; ++j) {
            float s = (j == 0) ? b_out[col] : 0.0f;
            #pragma unroll
            for (int nn = 0; nn < H; ++nn)
                s += cur[j][nn][p] * W_out[nn * 2 + col];
            outj[col][j][p] = s;
        }
    }
    __syncthreads();

    // ---------------- Phase 3: assemble u,v,p,f_u,f_v ----------------
    if (lane < TP) {
        int g = base + lane;
        if (g < n) {
            float P[NJ];
            #pragma unroll
            for (int j = 0; j < NJ; ++j) P[j] = outj[0][j][lane];   // psi jets
            float pv  = outj[1][0][lane];
            float p_x = outj[1][1][lane];
            float p_y = outj[1][2][lane];
            float l1 = lam1[0], l2 = lam2[0];
            float u =  P[2],  v = -P[1];                   // u=psi_y, v=-psi_x
            float u_t =  P[8],  u_x =  P[5],  u_y =  P[6];
            float u_xx = P[10], u_yy = P[12];
            float v_t = -P[7],  v_x = -P[4],  v_y = -P[5];
            float v_xx = -P[9], v_yy = -P[11];
            float f_u = u_t + l1 * (u * u_x + v * u_y) + p_x - l2 * (u_xx + u_yy);
            float f_v = v_t + l1 * (u * v_x + v * v_y) + p_y - l2 * (v_xx + v_yy);
            size_t N = (size_t)n;
            out[0 * N + g] = u;
            out[1 * N + g] = v;
            out[2 * N + g] = pv;
            out[3 * N + g] = f_u;
            out[4 * N + g] = f_v;
        }
    }
}

extern "C" void kernel_launch(void* const* d_in, const int* in_sizes, int n_in,
                              void* d_out, int out_size, void* d_ws, size_t ws_size,
                              hipStream_t stream) {
    (void)n_in; (void)out_size; (void)d_ws; (void)ws_size;
    const float* x     = (const float*)d_in[0];
    const float* y     = (const float*)d_in[1];
    const float* t     = (const float*)d_in[2];
    const float* W_in  = (const float*)d_in[3];
    const float* b_in  = (const float*)d_in[4];
    const float* W_hid = (const float*)d_in[5];
    const float* b_hid = (const float*)d_in[6];
    const float* W_out = (const float*)d_in[7];
    const float* b_out = (const float*)d_in[8];
    const float* lb    = (const float*)d_in[9];
    const float* ub    = (const float*)d_in[10];
    const float* l1    = (const float*)d_in[11];
    const float* l2    = (const float*)d_in[12];
    int n = in_sizes[0];
    int blocks = (n + TP - 1) / TP;
    pinn_ns_wmma<<<blocks, 32, 0, stream>>>(x, y, t, W_in, b_in, W_hid, b_hid,
                                            W_out, b_out, lb, ub, l1, l2,
                                            (float*)d_out, n);
}